// MultiScaleEncoder_13889924235807
// MI455X (gfx1250) — compile-verified
//
#include <hip/hip_runtime.h>
#include <hip/hip_bf16.h>
#include <stdint.h>

typedef __attribute__((ext_vector_type(4))) float v4f;
typedef __attribute__((ext_vector_type(4))) int   v4i;
typedef __attribute__((address_space(1))) v4i     gv4i;   // global int4 (async builtin)
typedef __attribute__((address_space(3))) v4i     lv4i;   // LDS int4   (async builtin)
typedef __attribute__((address_space(1))) v4f     gv4f;   // global float4 (gather)
typedef __attribute__((address_space(3))) v4f     lv4f;   // LDS float4    (gather)
typedef __attribute__((address_space(3))) void    lds_void;

#define THREADS   256
#define LDS_ROW0  87040            // OFFSETS[4]: first row of level 4
#define LDS_ROWS  341              // rows of levels 4..8 (256+64+16+4+1)
#define LDS_FLOATS (LDS_ROWS * 32) // 10912 floats = 43648 B

// Fetch one 16B feature chunk of a row. Hot levels (4..8) come from LDS via a
// true ds_load_b128 (AS3-typed pointer), cold levels from global (L2-resident).
// Distinct address spaces prevent the compiler from merging into a flat load.
__device__ __forceinline__ v4f fetch_row(const float* __restrict__ enc,
                                         const float* lds_f, int row, int c) {
  if (row >= LDS_ROW0) {
    return *(const lv4f*)(uintptr_t)(lds_f + (row - LDS_ROW0) * 32 + c);
  }
  return *(const gv4f*)(uintptr_t)(enc + (size_t)row * 32 + c);
}

// Bilinear interpolation of one level for a 4-float feature chunk.
__device__ __forceinline__ v4f interp_level(const float* __restrict__ enc,
                                            const float* lds_f,
                                            float cu, float cv, int level, int c) {
  int sh  = 8 - level;           // level_res = 256 >> level = 1 << sh
  int res = 1 << sh;
  // OFFSETS[l] = sum_{k<l} (256>>k)^2 = 4*(65536 - 65536>>2l)/3 (exact)
  int off = (int)(((65536u - (65536u >> (2 * level))) * 4u) / 3u);
  float rf = (float)res;
  float px = __fmaf_rn(cu, rf, -0.5f);
  float py = __fmaf_rn(cv, rf, -0.5f);
  float fx = floorf(px), fy = floorf(py);
  float wx = px - fx,    wy = py - fy;
  int ix = (int)fx, iy = (int)fy;       // in [-1, res-1]
  int m  = res - 1;                     // power-of-two wrap (handles -1 too)
  int x0 = ix & m, x1 = (ix + 1) & m;
  int y0 = iy & m, y1 = (iy + 1) & m;
  int r00 = (x0 << sh) + y0 + off;
  int r10 = (x1 << sh) + y0 + off;
  int r01 = (x0 << sh) + y1 + off;
  int r11 = (x1 << sh) + y1 + off;
  v4f f00 = fetch_row(enc, lds_f, r00, c);
  v4f f10 = fetch_row(enc, lds_f, r10, c);
  v4f f01 = fetch_row(enc, lds_f, r01, c);
  v4f f11 = fetch_row(enc, lds_f, r11, c);
  float w00 = (1.0f - wx) * (1.0f - wy);
  float w10 = wx * (1.0f - wy);
  float w01 = (1.0f - wx) * wy;
  float w11 = wx * wy;
  return f00 * w00 + f10 * w10 + f01 * w01 + f11 * w11;
}

__global__ __launch_bounds__(THREADS)
void MultiScaleEncoder_13889924235807_kernel(const float* __restrict__ x,
                                             const float* __restrict__ enc,
                                             float* __restrict__ out,
                                             int npoints) {
  __shared__ float lds_f[LDS_FLOATS];

  // ---- Stage hot levels (4..8, 43.6 KB) into LDS via async global->LDS ----
  {
    const v4f* gsrc = (const v4f*)(enc + (size_t)LDS_ROW0 * 32);
    for (int j = threadIdx.x; j < LDS_FLOATS / 4; j += THREADS) {
#if __has_builtin(__builtin_amdgcn_global_load_async_to_lds_b128)
      __builtin_amdgcn_global_load_async_to_lds_b128(
          (gv4i*)(uintptr_t)(gsrc + j),
          (lv4i*)(uintptr_t)(lds_f + 4 * j),
          0, 0);
#else
      asm volatile("global_load_async_to_lds_b128 %0, %1, off"
                   :
                   : "v"((lds_void*)(uintptr_t)(lds_f + 4 * j)),
                     "v"(gsrc + j)
                   : "memory");
#endif
    }
#if __has_builtin(__builtin_amdgcn_s_wait_asynccnt)
    __builtin_amdgcn_s_wait_asynccnt(0);
#else
    asm volatile("s_wait_asynccnt 0" ::: "memory");
#endif
    __syncthreads();
  }

  // ---- Main loop: 8 lanes per point, 4 features per lane ----
  int total  = npoints * 8;
  int stride = gridDim.x * THREADS;
  for (int i = blockIdx.x * THREADS + threadIdx.x; i < total; i += stride) {
    int p = i >> 3;
    int c = (i & 7) << 2;                       // feature chunk start
    const float* xp = x + (size_t)p * 3;
    float cu = __builtin_nontemporal_load(xp + 0);   // x is streamed once:
    float cv = __builtin_nontemporal_load(xp + 1);   // don't pollute L2,
    float du = __builtin_nontemporal_load(xp + 2);   // keep it for encoder

    float fp = fminf(fmaxf(2048.0f * du, 8.0f), 2048.0f);
    // searchsorted(STRIDES, fp, 'right') = floor(log2(fp)) - 2  (strides = 8*2^l)
    int e  = (int)(__float_as_uint(fp) >> 23) - 127;   // floor(log2(fp)), fp>=8
    int hi = e - 2;
    if (hi > 8) hi = 8;
    int lo = hi - 1;                                   // hi >= 1 always

    v4f vhi = interp_level(enc, lds_f, cu, cv, hi, c);
    v4f vlo = interp_level(enc, lds_f, cu, cv, lo, c);

    float fplo = (float)(8 << lo);                     // fp_hi - fp_lo == fp_lo
    float wl   = (fp - fplo) / fplo;
    v4f r = vlo + (vhi - vlo) * wl;

    // Streaming 256 MB output: non-temporal, fully coalesced (16 B/lane).
    __builtin_nontemporal_store(r, (v4f*)(out + (size_t)p * 32 + c));
  }
}

extern "C" void kernel_launch(void* const* d_in, const int* in_sizes, int n_in,
                              void* d_out, int out_size, void* d_ws, size_t ws_size,
                              hipStream_t stream) {
  const float* x   = (const float*)d_in[0];
  const float* enc = (const float*)d_in[1];
  float* out       = (float*)d_out;
  int npoints = in_sizes[0] / 3;

  int blocks = 4096;                       // persistent-style grid-stride
  long long total = (long long)npoints * 8;
  long long need  = (total + THREADS - 1) / THREADS;
  if (need < blocks) blocks = (int)(need > 0 ? need : 1);

  MultiScaleEncoder_13889924235807_kernel<<<blocks, THREADS, 0, stream>>>(
      x, enc, out, npoints);
}